// CombinedAdaptiveMambaLayer_64699387347590
// MI455X (gfx1250) — compile-verified
//
#include <hip/hip_runtime.h>
#include <math.h>

typedef float v2f __attribute__((ext_vector_type(2)));
typedef float v8f __attribute__((ext_vector_type(8)));

#define ACT_NONE 0
#define ACT_RELU 1
#define ACT_SOFTPLUS 2

static __host__ __device__ inline long cdivl(long a, long b) { return (a + b - 1) / b; }

#define WMMA4(ACC, Af, Bf) \
  (ACC) = __builtin_amdgcn_wmma_f32_16x16x4_f32(false, (Af), false, (Bf), (short)0, (ACC), false, false)

// ---------------------------------------------------------------------------
// Generic f32 WMMA GEMM:  C[m,n] = alpha * sum_k A[m,k]*B(k,n) + bias[n]
//   transB==0 : B given as W[N,K] row-major (PyTorch weight), B(k,n)=W[n*ldb+k]
//   transB==1 : B given as [K,N] row-major,                   B(k,n)=B[k*ldb+n]
// Composite batch: z = bb*nh + hh, per-operand (batch, head) strides.
// One wave computes a 32x16 C block (two 16x16 M-tiles sharing one B fragment)
// via V_WMMA_F32_16X16X4_F32 (exact f32 math). Inner loop is guard-free
// (v2f/global_load_b64 operand loads, software-pipelined); K remainder handled
// by a single guarded tail; out-of-range n handled by clamping the B pointer.
// Requires: M % 32 == 0, all strides even (8B alignment) -- true for all calls.
// ---------------------------------------------------------------------------
__global__ void wmma_gemm_f32(const float* __restrict__ A, const float* __restrict__ B,
                              const float* __restrict__ bias, float* __restrict__ C,
                              int M, int N, int K, int lda, int ldb, int ldc,
                              int nh, long sAb, long sAh, long sBb, long sBh,
                              long sCb, long sCh, float alpha, int act, int transB) {
  int z  = blockIdx.z;
  int bb = z / nh, hh = z % nh;
  const float* Ap = A + bb * sAb + hh * sAh;
  const float* Bp = B + bb * sBb + hh * sBh;
  float*       Cp = C + bb * sCb + hh * sCh;

  int wave = threadIdx.x >> 5;
  int lane = threadIdx.x & 31;
  int m0 = blockIdx.x * 32;                  // two 16-row tiles per wave
  int n0 = (blockIdx.y * 4 + wave) * 16;
  if (n0 >= N || m0 >= M) return;            // wave-uniform exit

  int r  = lane & 15;                        // A row within tile / B column within tile
  int kk = (lane >> 4) << 1;                 // K sub-offset 0 or 2 (ISA 16x4 layout)
  int n  = n0 + r;
  bool nok = n < N;
  int  nc  = nok ? n : (N - 1);              // clamp: loads in-bounds, result unstored

  v8f acc0 = {0.f, 0.f, 0.f, 0.f, 0.f, 0.f, 0.f, 0.f};
  v8f acc1 = {0.f, 0.f, 0.f, 0.f, 0.f, 0.f, 0.f, 0.f};

  const float* Arow0 = Ap + (long)(m0 + r) * lda + kk;
  const float* Arow1 = Arow0 + (long)16 * lda;
  int K4 = K & ~3;

  if (!transB) {
    const float* Brow = Bp + (long)nc * ldb + kk;
    v2f a0 = *(const v2f*)(Arow0);
    v2f a1 = *(const v2f*)(Arow1);
    v2f b  = *(const v2f*)(Brow);
    for (int k = 4; k < K4; k += 4) {
      v2f a0n = *(const v2f*)(Arow0 + k);
      v2f a1n = *(const v2f*)(Arow1 + k);
      v2f bn  = *(const v2f*)(Brow + k);
      WMMA4(acc0, a0, b);
      WMMA4(acc1, a1, b);
      a0 = a0n; a1 = a1n; b = bn;
    }
    WMMA4(acc0, a0, b);
    WMMA4(acc1, a1, b);
    if (K4 < K) {                            // guarded remainder (K=6 case)
      int ka = K4 + kk;
      v2f a0t, a1t, bt;
      a0t.x = (ka < K) ? Arow0[K4] : 0.f;  a0t.y = (ka + 1 < K) ? Arow0[K4 + 1] : 0.f;
      a1t.x = (ka < K) ? Arow1[K4] : 0.f;  a1t.y = (ka + 1 < K) ? Arow1[K4 + 1] : 0.f;
      bt.x  = (ka < K) ? Brow[K4]  : 0.f;  bt.y  = (ka + 1 < K) ? Brow[K4 + 1]  : 0.f;
      WMMA4(acc0, a0t, bt);
      WMMA4(acc1, a1t, bt);
    }
  } else {
    v2f a0 = *(const v2f*)(Arow0);
    v2f a1 = *(const v2f*)(Arow1);
    v2f b;
    b.x = Bp[(long)kk * ldb + nc];
    b.y = Bp[(long)(kk + 1) * ldb + nc];
    for (int k = 4; k < K4; k += 4) {
      v2f a0n = *(const v2f*)(Arow0 + k);
      v2f a1n = *(const v2f*)(Arow1 + k);
      v2f bn;
      bn.x = Bp[(long)(k + kk) * ldb + nc];
      bn.y = Bp[(long)(k + kk + 1) * ldb + nc];
      WMMA4(acc0, a0, b);
      WMMA4(acc1, a1, b);
      a0 = a0n; a1 = a1n; b = bn;
    }
    WMMA4(acc0, a0, b);
    WMMA4(acc1, a1, b);
    if (K4 < K) {
      int ka = K4 + kk;
      v2f a0t, a1t, bt;
      a0t.x = (ka < K) ? Arow0[K4] : 0.f;  a0t.y = (ka + 1 < K) ? Arow0[K4 + 1] : 0.f;
      a1t.x = (ka < K) ? Arow1[K4] : 0.f;  a1t.y = (ka + 1 < K) ? Arow1[K4 + 1] : 0.f;
      bt.x  = (ka < K) ? Bp[(long)ka * ldb + nc] : 0.f;
      bt.y  = (ka + 1 < K) ? Bp[(long)(ka + 1) * ldb + nc] : 0.f;
      WMMA4(acc0, a0t, bt);
      WMMA4(acc1, a1t, bt);
    }
  }

  if (nok) {
    float bv = (bias != nullptr) ? bias[n] : 0.f;
    int mb0 = m0 + ((lane >> 4) << 3);       // C layout: lanes 16-31 hold rows M=8..15
#pragma unroll
    for (int i = 0; i < 8; i++) {
      float v = acc0[i] * alpha + bv;
      if (act == ACT_RELU)          v = fmaxf(v, 0.f);
      else if (act == ACT_SOFTPLUS) v = (v > 20.f) ? v : log1pf(expf(v));
      Cp[(long)(mb0 + i) * ldc + n] = v;
    }
#pragma unroll
    for (int i = 0; i < 8; i++) {
      float v = acc1[i] * alpha + bv;
      if (act == ACT_RELU)          v = fmaxf(v, 0.f);
      else if (act == ACT_SOFTPLUS) v = (v > 20.f) ? v : log1pf(expf(v));
      Cp[(long)(mb0 + 16 + i) * ldc + n] = v;
    }
  }
}

// ---------------------------------------------------------------------------
// Stage 1: ViT helpers
// ---------------------------------------------------------------------------
__global__ void unfold_kernel(const float* __restrict__ x, float* __restrict__ pm, long total) {
  long i = (long)blockIdx.x * blockDim.x + threadIdx.x;
  if (i >= total) return;
  int  col = (int)(i % 6144);           // (c, kr, kc)
  long row = i / 6144;                  // b*256 + t
  int b = (int)(row >> 8), t = (int)(row & 255);
  int pi = t >> 4, pj = t & 15;
  int c = col >> 6, rem = col & 63, kr = rem >> 3, kc = rem & 7;
  pm[i] = x[(((long)b * 96 + c) * 128 + (pi * 8 + kr)) * 128 + pj * 8 + kc];
}

__device__ inline float tri_w(int i, float c, float scale) {
  float t = fabsf(((float)i - c) * scale);
  return (t < 1.f) ? (1.f - t) : 0.f;
}

// jax.image.resize 'bilinear' (antialias=True): triangle kernel widened by 28/16,
// half-pixel centers, per-dim weight renormalization.
__global__ void pe_resize_kernel(const float* __restrict__ pos, float* __restrict__ pe16) {
  int i = blockIdx.x * blockDim.x + threadIdx.x;
  if (i >= 256 * 96) return;
  int e = i % 96, t = i / 96;
  int orow = t >> 4, ocol = t & 15;
  const float scale = 16.f / 28.f, inv = 28.f / 16.f, rad = 1.75f;
  float cy = (orow + 0.5f) * inv - 0.5f;
  float cx = (ocol + 0.5f) * inv - 0.5f;
  int y0 = (int)ceilf(cy - rad); if (y0 < 0) y0 = 0;
  int y1 = (int)floorf(cy + rad); if (y1 > 27) y1 = 27;
  int x0 = (int)ceilf(cx - rad); if (x0 < 0) x0 = 0;
  int x1 = (int)floorf(cx + rad); if (x1 > 27) x1 = 27;
  float swy = 0.f, swx = 0.f;
  for (int y = y0; y <= y1; y++) swy += tri_w(y, cy, scale);
  for (int xq = x0; xq <= x1; xq++) swx += tri_w(xq, cx, scale);
  float acc = 0.f;
  for (int y = y0; y <= y1; y++) {
    float wy = tri_w(y, cy, scale) / swy;
    for (int xq = x0; xq <= x1; xq++) {
      float wx = tri_w(xq, cx, scale) / swx;
      acc += wy * wx * pos[((long)y * 28 + xq) * 96 + e];
    }
  }
  pe16[(long)t * 96 + e] = acc;
}

__global__ void add_pe_kernel(float* __restrict__ patches, const float* __restrict__ pe16, int total) {
  int i = blockIdx.x * blockDim.x + threadIdx.x;
  if (i >= total) return;
  int e = i % 96;
  int t = (i / 96) % 256;
  patches[i] += 0.1f * pe16[t * 96 + e];
}

__global__ void softmax256_kernel(float* __restrict__ S, int rows) {
  int row = blockIdx.x * (blockDim.x >> 5) + (threadIdx.x >> 5);
  if (row >= rows) return;
  int lane = threadIdx.x & 31;
  float* p = S + (long)row * 256;
  float v[8];
  float mx = -3e38f;
#pragma unroll
  for (int j = 0; j < 8; j++) { v[j] = p[lane + j * 32]; mx = fmaxf(mx, v[j]); }
  for (int m = 16; m; m >>= 1) mx = fmaxf(mx, __shfl_xor(mx, m, 32));
  float sum = 0.f;
#pragma unroll
  for (int j = 0; j < 8; j++) { v[j] = expf(v[j] - mx); sum += v[j]; }
  for (int m = 16; m; m >>= 1) sum += __shfl_xor(sum, m, 32);
  float r = 1.f / sum;
#pragma unroll
  for (int j = 0; j < 8; j++) p[lane + j * 32] = v[j] * r;
}

// out = LayerNorm(a + b) over 96 features, one wave per row
__global__ void ln_residual_kernel(const float* __restrict__ a, const float* __restrict__ bsrc,
                                   const float* __restrict__ w, const float* __restrict__ bi,
                                   float* __restrict__ out, int rows) {
  int row = blockIdx.x * (blockDim.x >> 5) + (threadIdx.x >> 5);
  if (row >= rows) return;
  int lane = threadIdx.x & 31;
  float v[3];
#pragma unroll
  for (int j = 0; j < 3; j++) {
    int e = lane + j * 32;
    v[j] = a[(long)row * 96 + e] + bsrc[(long)row * 96 + e];
  }
  float s = v[0] + v[1] + v[2];
  for (int m = 16; m; m >>= 1) s += __shfl_xor(s, m, 32);
  float mu = s / 96.f;
  float q = 0.f;
#pragma unroll
  for (int j = 0; j < 3; j++) { float d = v[j] - mu; q += d * d; }
  for (int m = 16; m; m >>= 1) q += __shfl_xor(q, m, 32);
  float rstd = rsqrtf(q / 96.f + 1e-5f);
#pragma unroll
  for (int j = 0; j < 3; j++) {
    int e = lane + j * 32;
    out[(long)row * 96 + e] = (v[j] - mu) * rstd * w[e] + bi[e];
  }
}

__device__ inline float sigm(float x) { return 1.f / (1.f + expf(-x)); }

__global__ void score_kernel(const float* __restrict__ src2, const float* __restrict__ patches,
                             const float* __restrict__ sw, const float* __restrict__ sb,
                             float* __restrict__ smap, int rows) {
  int row = blockIdx.x * (blockDim.x >> 5) + (threadIdx.x >> 5);
  if (row >= rows) return;
  int lane = threadIdx.x & 31;
  float d = 0.f, p[3];
#pragma unroll
  for (int j = 0; j < 3; j++) {
    int e = lane + j * 32;
    d += src2[(long)row * 96 + e] * sw[e];
    p[j] = patches[(long)row * 96 + e];
  }
  for (int m = 16; m; m >>= 1) d += __shfl_xor(d, m, 32);
  float s = p[0] + p[1] + p[2];
  for (int m = 16; m; m >>= 1) s += __shfl_xor(s, m, 32);
  float mu = s / 96.f;
  float q = 0.f;
#pragma unroll
  for (int j = 0; j < 3; j++) { float dd = p[j] - mu; q += dd * dd; }
  for (int m = 16; m; m >>= 1) q += __shfl_xor(q, m, 32);
  float pvar = q / 95.f;                       // torch.var unbiased
  float fin = 0.7f * sigm(d + sb[0]) + 0.3f * sigm(pvar * 10.f);
  if (lane == 0) smap[row] = fin;
}

// ---------------------------------------------------------------------------
// Stage 2: greedy traversal, one wave per batch (first-max tie-break argmax)
// ---------------------------------------------------------------------------
__global__ void traversal_kernel(const float* __restrict__ score, int* __restrict__ order) {
  int b = blockIdx.x;
  const float* s = score + (long)b * 256;
  __shared__ float ls[256];
  __shared__ unsigned vis[8];
  int lane = threadIdx.x;
  for (int i = lane; i < 256; i += 32) ls[i] = s[i];
  if (lane < 8) vis[lane] = 0u;
  __syncthreads();

  const int di[4] = {-1, 1, 0, 0};
  const int dj[4] = {0, 0, -1, 1};

  int cur = 0;
  for (int stp = 0; stp < 256; stp++) {
    int nxt;
    bool need_jump = true;
    if (stp > 0) {
      int ci = cur >> 4, cj = cur & 15;
      int best = -1; float bvn = -3e38f;
      for (int k = 0; k < 4; k++) {
        int ni = ci + di[k], nj = cj + dj[k];
        if (ni < 0 || ni >= 16 || nj < 0 || nj >= 16) continue;
        int nidx = ni * 16 + nj;
        if ((vis[nidx >> 5] >> (nidx & 31)) & 1u) continue;
        float v = ls[nidx];
        if (v > bvn) { bvn = v; best = nidx; }
      }
      if (best >= 0) { nxt = best; need_jump = false; }
    }
    if (need_jump) {                         // global argmax over unvisited
      float bv = -3e38f; int bi = lane * 8;
#pragma unroll
      for (int j = 0; j < 8; j++) {
        int i = lane * 8 + j;
        float v = ((vis[i >> 5] >> (i & 31)) & 1u) ? -3e38f : ls[i];
        if (v > bv) { bv = v; bi = i; }
      }
      for (int m = 16; m >= 1; m >>= 1) {
        float ov = __shfl_xor(bv, m, 32);
        int   oi = __shfl_xor(bi, m, 32);
        if (ov > bv || (ov == bv && oi < bi)) { bv = ov; bi = oi; }
      }
      nxt = bi;
    }
    if (lane == 0) {
      vis[nxt >> 5] |= 1u << (nxt & 31);
      order[(long)b * 256 + stp] = nxt;
    }
    __syncthreads();
    cur = nxt;
  }
}

__global__ void build_idx_kernel(const int* __restrict__ order, int* __restrict__ idx) {
  int i = blockIdx.x * blockDim.x + threadIdx.x;
  if (i >= 2 * 16384) return;
  int b = i >> 14, li = i & 16383;
  int t = li >> 6, o = li & 63;
  int p = order[b * 256 + t];
  int pi = p >> 4, pj = p & 15;
  idx[i] = (pi * 8 + (o >> 3)) * 128 + pj * 8 + (o & 7);
}

// tokens = LN(x_gather + pe_gather), one wave per token
__global__ void tokens_kernel(const float* __restrict__ x, const float* __restrict__ pe,
                              const int* __restrict__ idx, const float* __restrict__ nw,
                              const float* __restrict__ nb, float* __restrict__ tok) {
  int row = blockIdx.x * (blockDim.x >> 5) + (threadIdx.x >> 5);
  if (row >= 2 * 16384) return;
  int lane = threadIdx.x & 31;
  int b = row >> 14;
  int pix = idx[row];
  float v[3];
#pragma unroll
  for (int j = 0; j < 3; j++) {
    int c = lane + j * 32;
    v[j] = x[((long)b * 96 + c) * 16384 + pix] + pe[((long)b * 16384 + pix) * 96 + c];
  }
  float s = v[0] + v[1] + v[2];
  for (int m = 16; m; m >>= 1) s += __shfl_xor(s, m, 32);
  float mu = s / 96.f;
  float q = 0.f;
#pragma unroll
  for (int j = 0; j < 3; j++) { float d = v[j] - mu; q += d * d; }
  for (int m = 16; m; m >>= 1) q += __shfl_xor(q, m, 32);
  float rstd = rsqrtf(q / 96.f + 1e-5f);
#pragma unroll
  for (int j = 0; j < 3; j++) {
    int c = lane + j * 32;
    tok[(long)row * 96 + c] = (v[j] - mu) * rstd * nw[c] + nb[c];
  }
}

// ---------------------------------------------------------------------------
// Stage 3: Mamba
// ---------------------------------------------------------------------------
__global__ void conv_silu_kernel(const float* __restrict__ xz, const float* __restrict__ cw,
                                 const float* __restrict__ cb, float* __restrict__ xc, long total) {
  long i = (long)blockIdx.x * blockDim.x + threadIdx.x;
  if (i >= total) return;
  int d = (int)(i % 192);
  long bl = i / 192;
  int l = (int)(bl & 16383), b = (int)(bl >> 14);
  const float* w = cw + d * 4;
  float acc = cb[d];
#pragma unroll
  for (int j = 0; j < 4; j++) {
    int ls = l - 3 + j;
    if (ls >= 0) acc += w[j] * xz[((long)b * 16384 + ls) * 384 + d];
  }
  xc[i] = acc / (1.f + expf(-acc));           // SiLU
}

// Selective scan: half-wave (16 lanes) per (b,d) channel, lane = state s.
// 24 blocks x 256 threads cover B*192 channels; 16384 sequential steps each.
__global__ void scan_kernel(const float* __restrict__ dt, const float* __restrict__ xc,
                            const float* __restrict__ dbl, const float* __restrict__ A_log,
                            float* __restrict__ ys) {
  int blk = blockIdx.x;
  int b = blk / 12, d0 = (blk % 12) * 16;
  int wave = threadIdx.x >> 5, lane = threadIdx.x & 31;
  int d = d0 + wave * 2 + (lane >> 4);
  int s = lane & 15;
  float Ads = -expf(A_log[d * 16 + s]);
  float h = 0.f;
  const float* dtb = dt  + (long)b * 16384 * 192;
  const float* xb  = xc  + (long)b * 16384 * 192;
  const float* db  = dbl + (long)b * 16384 * 38;
  float*       yb  = ys  + (long)b * 16384 * 192;
  for (int l = 0; l < 16384; l++) {
    float dtv = dtb[(long)l * 192 + d];
    float xv  = xb[(long)l * 192 + d];
    float Bv  = db[(long)l * 38 + 6 + s];
    float Cv  = db[(long)l * 38 + 22 + s];
    h = expf(dtv * Ads) * h + dtv * xv * Bv;
    float p = h * Cv;
    p += __shfl_xor(p, 1, 32);
    p += __shfl_xor(p, 2, 32);
    p += __shfl_xor(p, 4, 32);
    p += __shfl_xor(p, 8, 32);                // reduction stays in 16-lane half
    if (s == 0) yb[(long)l * 192 + d] = p;
  }
}

__global__ void yact_kernel(float* __restrict__ ys, const float* __restrict__ xc,
                            const float* __restrict__ xz, const float* __restrict__ Dsk, long total) {
  long i = (long)blockIdx.x * blockDim.x + threadIdx.x;
  if (i >= total) return;
  int d = (int)(i % 192);
  long bl = i / 192;
  float v = ys[i] + xc[i] * Dsk[d];
  float z = xz[bl * 384 + 192 + d];
  ys[i] = v * (z / (1.f + expf(-z)));
}

__global__ void zero_kernel(float* __restrict__ p, long n) {
  long i = (long)blockIdx.x * blockDim.x + threadIdx.x;
  if (i < n) p[i] = 0.f;
}

__global__ void scatter_kernel(const float* __restrict__ yout, const int* __restrict__ idx,
                               float* __restrict__ out, long total) {
  long i = (long)blockIdx.x * blockDim.x + threadIdx.x;
  if (i >= total) return;
  int c = (int)(i % 96);
  long bl = i / 96;
  int l = (int)(bl & 16383), b = (int)(bl >> 14);
  int pix = idx[(long)b * 16384 + l];
  out[((long)b * 96 + c) * 16384 + pix] = yout[i];
}

// ---------------------------------------------------------------------------
extern "C" void kernel_launch(void* const* d_in, const int* in_sizes, int n_in,
                              void* d_out, int out_size, void* d_ws, size_t ws_size,
                              hipStream_t stream) {
  const float* x          = (const float*)d_in[0];
  const float* pe         = (const float*)d_in[1];
  const float* pos_embed  = (const float*)d_in[2];
  const float* vit_proj_w = (const float*)d_in[3];
  const float* vit_proj_b = (const float*)d_in[4];
  const float* attn_in_w  = (const float*)d_in[5];
  const float* attn_in_b  = (const float*)d_in[6];
  const float* attn_out_w = (const float*)d_in[7];
  const float* attn_out_b = (const float*)d_in[8];
  const float* ln1_w      = (const float*)d_in[9];
  const float* ln1_b      = (const float*)d_in[10];
  const float* ff1_w      = (const float*)d_in[11];
  const float* ff1_b      = (const float*)d_in[12];
  const float* ff2_w      = (const float*)d_in[13];
  const float* ff2_b      = (const float*)d_in[14];
  const float* ln2_w      = (const float*)d_in[15];
  const float* ln2_b      = (const float*)d_in[16];
  const float* score_w    = (const float*)d_in[17];
  const float* score_b    = (const float*)d_in[18];
  const float* norm_w     = (const float*)d_in[19];
  const float* norm_b     = (const float*)d_in[20];
  const float* in_proj_w  = (const float*)d_in[21];
  const float* conv_w     = (const float*)d_in[22];
  const float* conv_b     = (const float*)d_in[23];
  const float* x_proj_w   = (const float*)d_in[24];
  const float* dt_w       = (const float*)d_in[25];
  const float* dt_b       = (const float*)d_in[26];
  const float* A_log      = (const float*)d_in[27];
  const float* D_skip     = (const float*)d_in[28];
  const float* out_proj_w = (const float*)d_in[29];
  (void)in_sizes; (void)n_in; (void)out_size; (void)ws_size;

  float* ws = (float*)d_ws;
  size_t o = 0;
  float* pm      = ws + o; o += 3145728;   // unfold 512x6144; reused as yout later
  float* patches = ws + o; o += 49152;     // 512x96
  float* pe16    = ws + o; o += 24576;     // 256x96
  float* qkv     = ws + o; o += 147456;    // 512x288
  float* Smat    = ws + o; o += 262144;    // 2x2x256x256
  float* ctx     = ws + o; o += 49152;     // 512x96
  float* tmp     = ws + o; o += 49152;     // 512x96 (attn-out, then ff2)
  float* src     = ws + o; o += 49152;
  float* src2    = ws + o; o += 49152;
  float* ffb     = ws + o; o += 1048576;   // 512x2048
  float* smap    = ws + o; o += 512;
  float* tok     = ws + o; o += 3145728;   // 2x16384x96
  float* xz      = ws + o; o += 12582912;  // 2x16384x384
  float* xc      = ws + o; o += 6291456;   // 2x16384x192
  float* dbl     = ws + o; o += 1245184;   // 2x16384x38
  float* dtb     = ws + o; o += 6291456;   // 2x16384x192
  float* ysb     = ws + o; o += 6291456;   // 2x16384x192
  int* order = (int*)(ws + o); o += 512;
  int* idxb  = (int*)(ws + o); o += 32768;
  float* yout = pm;                         // reuse unfold buffer (same size)
  float* out  = (float*)d_out;

  // ---- ViT score map ----
  { long tot = 512L * 6144; unfold_kernel<<<(unsigned)cdivl(tot, 256), 256, 0, stream>>>(x, pm, tot); }
  pe_resize_kernel<<<(unsigned)cdivl(24576, 256), 256, 0, stream>>>(pos_embed, pe16);
  wmma_gemm_f32<<<dim3(16, 2, 1), 128, 0, stream>>>(pm, vit_proj_w, vit_proj_b, patches,
      512, 96, 6144, 6144, 6144, 96, 1, 0, 0, 0, 0, 0, 0, 1.f, ACT_NONE, 0);
  add_pe_kernel<<<(unsigned)cdivl(49152, 256), 256, 0, stream>>>(patches, pe16, 49152);
  wmma_gemm_f32<<<dim3(16, 5, 1), 128, 0, stream>>>(patches, attn_in_w, attn_in_b, qkv,
      512, 288, 96, 96, 96, 288, 1, 0, 0, 0, 0, 0, 0, 1.f, ACT_NONE, 0);
  // attention scores: q @ k^T / sqrt(48), batched over (b,h)
  wmma_gemm_f32<<<dim3(8, 4, 4), 128, 0, stream>>>(qkv, qkv + 96, nullptr, Smat,
      256, 256, 48, 288, 288, 256, 2, 256L * 288, 48, 256L * 288, 48, 131072, 65536,
      1.0f / sqrtf(48.f), ACT_NONE, 0);
  softmax256_kernel<<<256, 128, 0, stream>>>(Smat, 1024);
  // ctx = attn @ v  (B stored [K,N] -> transB=1)
  wmma_gemm_f32<<<dim3(8, 1, 4), 128, 0, stream>>>(Smat, qkv + 192, nullptr, ctx,
      256, 48, 256, 256, 288, 96, 2, 131072, 65536, 256L * 288, 48, 256L * 96, 48,
      1.f, ACT_NONE, 1);
  wmma_gemm_f32<<<dim3(16, 2, 1), 128, 0, stream>>>(ctx, attn_out_w, attn_out_b, tmp,
      512, 96, 96, 96, 96, 96, 1, 0, 0, 0, 0, 0, 0, 1.f, ACT_NONE, 0);
  ln_residual_kernel<<<128, 128, 0, stream>>>(patches, tmp, ln1_w, ln1_b, src, 512);
  wmma_gemm_f32<<<dim3(16, 32, 1), 128, 0, stream>>>(src, ff1_w, ff1_b, ffb,
      512, 2048, 96, 96, 96, 2048, 1, 0, 0, 0, 0, 0, 0, 1.f, ACT_RELU, 0);
  wmma_gemm_f32<<<dim3(16, 2, 1), 128, 0, stream>>>(ffb, ff2_w, ff2_b, tmp,
      512, 96, 2048, 2048, 2048, 96, 1, 0, 0, 0, 0, 0, 0, 1.f, ACT_NONE, 0);
  ln_residual_kernel<<<128, 128, 0, stream>>>(src, tmp, ln2_w, ln2_b, src2, 512);
  score_kernel<<<128, 128, 0, stream>>>(src2, patches, score_w, score_b, smap, 512);

  // ---- traversal + gather ----
  traversal_kernel<<<2, 32, 0, stream>>>(smap, order);
  build_idx_kernel<<<(unsigned)cdivl(32768, 256), 256, 0, stream>>>(order, idxb);
  tokens_kernel<<<8192, 128, 0, stream>>>(x, pe, idxb, norm_w, norm_b, tok);

  // ---- Mamba ----
  wmma_gemm_f32<<<dim3(1024, 6, 1), 128, 0, stream>>>(tok, in_proj_w, nullptr, xz,
      32768, 384, 96, 96, 96, 384, 1, 0, 0, 0, 0, 0, 0, 1.f, ACT_NONE, 0);
  conv_silu_kernel<<<(unsigned)cdivl(6291456L, 256), 256, 0, stream>>>(xz, conv_w, conv_b, xc, 6291456L);
  wmma_gemm_f32<<<dim3(1024, 1, 1), 128, 0, stream>>>(xc, x_proj_w, nullptr, dbl,
      32768, 38, 192, 192, 192, 38, 1, 0, 0, 0, 0, 0, 0, 1.f, ACT_NONE, 0);
  wmma_gemm_f32<<<dim3(1024, 3, 1), 128, 0, stream>>>(dbl, dt_w, dt_b, dtb,
      32768, 192, 6, 38, 6, 192, 1, 0, 0, 0, 0, 0, 0, 1.f, ACT_SOFTPLUS, 0);
  scan_kernel<<<24, 256, 0, stream>>>(dtb, xc, dbl, A_log, ysb);
  yact_kernel<<<(unsigned)cdivl(6291456L, 256), 256, 0, stream>>>(ysb, xc, xz, D_skip, 6291456L);
  wmma_gemm_f32<<<dim3(1024, 2, 1), 128, 0, stream>>>(ysb, out_proj_w, nullptr, yout,
      32768, 96, 192, 192, 192, 96, 1, 0, 0, 0, 0, 0, 0, 1.f, ACT_NONE, 0);

  // ---- scatter back to raster order ----
  zero_kernel<<<(unsigned)cdivl(3145728L, 256), 256, 0, stream>>>(out, 3145728L);
  scatter_kernel<<<(unsigned)cdivl(3145728L, 256), 256, 0, stream>>>(yout, idxb, out, 3145728L);
}